// DotProductAttention_55001351192898
// MI455X (gfx1250) — compile-verified
//
#include <hip/hip_runtime.h>

typedef float v2f __attribute__((ext_vector_type(2)));
typedef float v8f __attribute__((ext_vector_type(8)));

#define SEQ 2048
#define HD  64
#define NB  16
#define SCALE 0.125f   // 1/sqrt(64)

// ---------------------------------------------------------------------------
// Pass A: per-key column stats of the (query-axis) softmax.
// One wave handles one (batch, 16-key tile); streams all query tiles.
// S_tile[16q x 16k] = Q_tile(16x64) * K_tile(16x64)^T via 16x v_wmma_f32_16x16x4_f32
// ---------------------------------------------------------------------------
__global__ __launch_bounds__(128) void attn_colstats_kernel(
    const float* __restrict__ q, const float* __restrict__ k,
    float* __restrict__ mbuf, float* __restrict__ lbuf) {
  const int lane = threadIdx.x & 31;
  const int wid  = blockIdx.x * 4 + (threadIdx.x >> 5);
  const int b    = wid >> 7;        // 128 key tiles per batch
  const int kt   = wid & 127;
  const int row  = lane & 15;
  const int h    = lane >> 4;       // half-wave selector

  // K tile fragments (B operand): lane holds K-matrix row `row`, elems {4kk+2h, +1}
  const float* ktile = k + ((size_t)b * SEQ + (size_t)kt * 16) * HD;
  v2f kf[16];
#pragma unroll
  for (int kk = 0; kk < 16; ++kk)
    kf[kk] = *(const v2f*)(ktile + row * HD + 4 * kk + 2 * h);

  float runmax = -3.0e38f;
  float runsum = 0.0f;

  const float* qb = q + (size_t)b * SEQ * HD;
  for (int qt = 0; qt < SEQ / 16; ++qt) {
    const float* qtile = qb + (size_t)qt * 16 * HD;
    v2f qf[16];
#pragma unroll
    for (int kk = 0; kk < 16; ++kk)
      qf[kk] = *(const v2f*)(qtile + row * HD + 4 * kk + 2 * h);

    v8f acc = {};
#pragma unroll
    for (int kk = 0; kk < 16; ++kk)
      acc = __builtin_amdgcn_wmma_f32_16x16x4_f32(false, qf[kk], false, kf[kk],
                                                  (short)0, acc, false, false);
    // acc[i] = score(row M = i + 8h, key col N = row) -- all same key column per lane
    float tmax = runmax;
#pragma unroll
    for (int i = 0; i < 8; ++i) tmax = fmaxf(tmax, acc[i] * SCALE);
    float s = runsum * __expf(runmax - tmax);
#pragma unroll
    for (int i = 0; i < 8; ++i) s += __expf(acc[i] * SCALE - tmax);
    runmax = tmax;
    runsum = s;
  }

  // merge the two row-halves (lane ^ 16 owns the same key column, rows 8..15)
  float om = __shfl_xor(runmax, 16, 32);
  float os = __shfl_xor(runsum, 16, 32);
  float m  = fmaxf(runmax, om);
  float ls = runsum * __expf(runmax - m) + os * __expf(om - m);
  if (lane < 16) {
    mbuf[(size_t)b * SEQ + kt * 16 + lane] = m;
    lbuf[(size_t)b * SEQ + kt * 16 + lane] = ls;
  }
}

// ---------------------------------------------------------------------------
// Pass B: out[16q x 64d] = sum_ktiles  exp(S - m)/l  @  V_tile(16x64)
// One wave per (batch, 16-query tile). P relayout (C/D-frag -> A-frag)
// goes through a per-wave padded LDS tile; same-wave DS ops are in-order.
// ---------------------------------------------------------------------------
__global__ __launch_bounds__(128) void attn_out_kernel(
    const float* __restrict__ q, const float* __restrict__ k,
    const float* __restrict__ v, const float* __restrict__ mbuf,
    const float* __restrict__ lbuf, float* __restrict__ out) {
  __shared__ float pshare[4][16 * 17];   // stride-17 pad: conflict-free strided reads

  const int lane  = threadIdx.x & 31;
  const int wslot = threadIdx.x >> 5;
  const int wid   = blockIdx.x * 4 + wslot;
  const int b     = wid >> 7;            // 128 query tiles per batch
  const int qt    = wid & 127;
  const int row   = lane & 15;
  const int h     = lane >> 4;

  float* P = &pshare[wslot][0];

  const float* qtile = q + ((size_t)b * SEQ + (size_t)qt * 16) * HD;
  v2f qf[16];
#pragma unroll
  for (int kk = 0; kk < 16; ++kk)
    qf[kk] = *(const v2f*)(qtile + row * HD + 4 * kk + 2 * h);

  v8f oacc[4] = {{}, {}, {}, {}};        // 16q x 64d output accumulator

  const float* kb = k + (size_t)b * SEQ * HD;
  const float* vb = v + (size_t)b * SEQ * HD;
  const float* mb = mbuf + (size_t)b * SEQ;
  const float* lb = lbuf + (size_t)b * SEQ;

  for (int kt = 0; kt < SEQ / 16; ++kt) {
    // ---- scores: Q_tile * K_tile^T ----
    const float* ktile = kb + (size_t)kt * 16 * HD;
    v2f kf[16];
#pragma unroll
    for (int kk = 0; kk < 16; ++kk)
      kf[kk] = *(const v2f*)(ktile + row * HD + 4 * kk + 2 * h);

    v8f acc = {};
#pragma unroll
    for (int kk = 0; kk < 16; ++kk)
      acc = __builtin_amdgcn_wmma_f32_16x16x4_f32(false, qf[kk], false, kf[kk],
                                                  (short)0, acc, false, false);

    // ---- normalize by column stats; stash P in LDS (row M = i+8h, col N = row) ----
    const int   kcol = kt * 16 + row;
    const float mval = mb[kcol];
    const float linv = 1.0f / lb[kcol];
#pragma unroll
    for (int i = 0; i < 8; ++i) {
      float p = __expf(acc[i] * SCALE - mval) * linv;
      P[(i + 8 * h) * 17 + row] = p;
    }

    // ---- P(16x16) @ V_tile(16x64): 4 d-chunks x 4 k-steps of f32 WMMA ----
#pragma unroll
    for (int c = 0; c < 4; ++c) {
#pragma unroll
      for (int kk = 0; kk < 4; ++kk) {
        v2f pa = { P[row * 17 + 4 * kk + 2 * h],
                   P[row * 17 + 4 * kk + 2 * h + 1] };
        const float* vrow = vb + ((size_t)kt * 16 + 4 * kk + 2 * h) * HD + c * 16 + row;
        v2f vf = { vrow[0], vrow[HD] };
        oacc[c] = __builtin_amdgcn_wmma_f32_16x16x4_f32(false, pa, false, vf,
                                                        (short)0, oacc[c], false, false);
      }
    }
  }

  // ---- store: lane owns d-col (c*16 + row), rows i + 8h ----
  float* ob = out + ((size_t)b * SEQ + (size_t)qt * 16) * HD;
#pragma unroll
  for (int c = 0; c < 4; ++c)
#pragma unroll
    for (int i = 0; i < 8; ++i)
      ob[(i + 8 * h) * HD + c * 16 + row] = oacc[c][i];
}

extern "C" void kernel_launch(void* const* d_in, const int* in_sizes, int n_in,
                              void* d_out, int out_size, void* d_ws, size_t ws_size,
                              hipStream_t stream) {
  const float* q = (const float*)d_in[0];
  const float* k = (const float*)d_in[1];
  const float* v = (const float*)d_in[2];
  float* out  = (float*)d_out;
  float* mbuf = (float*)d_ws;               // NB*SEQ floats
  float* lbuf = mbuf + (size_t)NB * SEQ;    // NB*SEQ floats (256 KB total < ws)

  dim3 blk(128);                            // 4 wave32 per block
  dim3 grid((NB * (SEQ / 16)) / 4);         // 2048 waves -> 512 blocks

  attn_colstats_kernel<<<grid, blk, 0, stream>>>(q, k, mbuf, lbuf);
  attn_out_kernel<<<grid, blk, 0, stream>>>(q, k, v, mbuf, lbuf, out);
}